// TransNetImplicit_32573031973311
// MI455X (gfx1250) — compile-verified
//
#include <hip/hip_runtime.h>
#include <math.h>

// ---------------------------------------------------------------------------
// TransNet implicit layers on MI455X (gfx1250), fp32 WMMA path.
//   v0' = -Minv^T v1 ; v1' = Minv (v0 + q)   with Minv = B0^-1
// Batch-major (32768 x 256 row-major): each step is a plain GEMM
//   C(M,N) = A(M,K) @ B(K,N) [+ bias[n]] [tanh side-output]
// Memory-bound (~0.9 GB @ 23.3 TB/s). fp32 WMMA 16x16x4 for accuracy through
// 6 chained solves. A-operand staged with GLOBAL_LOAD_ASYNC_TO_LDS_B128
// (ASYNCcnt) into double-buffered LDS so the DMA of tile k+1 overlaps the
// WMMAs of tile k with a single barrier per K-tile.
// ---------------------------------------------------------------------------

typedef __attribute__((ext_vector_type(2))) float v2f;
typedef __attribute__((ext_vector_type(4))) float v4f;
typedef __attribute__((ext_vector_type(8))) float v8f;
typedef int b128_t __attribute__((vector_size(4 * sizeof(int))));  // builtin's pointee

#define LAYERS 6
#define UDIM   256
#define DIN    784
#define DOUT   10
#define BATCH  32768

constexpr int KB  = 16;      // K-extent staged in LDS per step (4 wmma k-steps)
constexpr int KBP = KB + 4;  // padded row: 20 floats = 80B (16B-aligned rows)

#define AS1 __attribute__((address_space(1)))
#define AS3 __attribute__((address_space(3)))

// CDNA5 async global->LDS copy (tracked by ASYNCcnt). Guarded so the register
// fallback path still compiles if the toolchain spells the builtin otherwise.
#if __has_builtin(__builtin_amdgcn_global_load_async_to_lds_b128)
#define USE_ASYNC_LDS 1
#else
#define USE_ASYNC_LDS 0
#endif

#if __has_builtin(__builtin_amdgcn_s_wait_asynccnt)
#define WAIT_ASYNC_0() __builtin_amdgcn_s_wait_asynccnt(0)
#else
#define WAIT_ASYNC_0() asm volatile("s_wait_asynccnt 0x0" ::: "memory")
#endif

// ---------------------------------------------------------------------------
// Gauss-Jordan inversion with partial pivoting, one workgroup per layer.
// Augmented [B0 | I] (256 x 512) lives in workspace (L2-resident).
// Thread t owns columns {t, t+256}; eliminations are fully coalesced.
// Epilogue writes Minv^T (B-operand for v1 GEMM), -Minv (B-operand for v0
// GEMM) and cvec = Minv @ q (folded bias for the v1 GEMM).
// ---------------------------------------------------------------------------
__global__ __launch_bounds__(256) void gj_invert_kernel(
    const float* __restrict__ B0,    // [L][256][256]
    const float* __restrict__ q,     // [L][256]
    float* __restrict__ aug,         // [L][256][512] scratch
    float* __restrict__ T,           // [L][256][256]  Minv^T
    float* __restrict__ Nn,          // [L][256][256]  -Minv
    float* __restrict__ cvec)        // [L][256]       Minv @ q
{
  const int l   = blockIdx.x;
  const int tid = threadIdx.x;
  float* W = aug + (size_t)l * UDIM * (2 * UDIM);

  __shared__ float rowbuf[2 * UDIM];
  __shared__ float colbuf[UDIM];
  __shared__ float redv[UDIM];
  __shared__ int   redi[UDIM];

  // init augmented matrix [B0_l | I]
  for (int idx = tid; idx < UDIM * 2 * UDIM; idx += 256) {
    int r = idx >> 9, c = idx & 511;
    float v;
    if (c < UDIM) v = B0[(size_t)l * UDIM * UDIM + (size_t)r * UDIM + c];
    else          v = (c - UDIM == r) ? 1.0f : 0.0f;
    W[idx] = v;
  }
  __syncthreads();

  for (int k = 0; k < UDIM; ++k) {
    // ---- partial pivot: argmax_{r>=k} |W[r][k]| ----
    redv[tid] = (tid >= k) ? fabsf(W[(size_t)tid * 2 * UDIM + k]) : -1.0f;
    redi[tid] = tid;
    __syncthreads();
    for (int s = 128; s > 0; s >>= 1) {
      if (tid < s && redv[tid + s] > redv[tid]) {
        redv[tid] = redv[tid + s];
        redi[tid] = redi[tid + s];
      }
      __syncthreads();
    }
    const int   p    = redi[0];
    const float pinv = 1.0f / W[(size_t)p * 2 * UDIM + k];
    __syncthreads();

    // ---- swap rows k<->p, normalize pivot row, broadcast it ----
    for (int c = tid; c < 2 * UDIM; c += 256) {
      float wkc = W[(size_t)p * 2 * UDIM + c];  // value moving into row k
      float wpc = W[(size_t)k * 2 * UDIM + c];  // value moving into row p
      float nk  = wkc * pinv;
      if (p != k) W[(size_t)p * 2 * UDIM + c] = wpc;
      W[(size_t)k * 2 * UDIM + c] = nk;
      rowbuf[c] = nk;
    }
    __syncthreads();
    colbuf[tid] = W[(size_t)tid * 2 * UDIM + k];  // multipliers (post-swap)
    __syncthreads();

    // ---- eliminate column k from every other row (coalesced over c) ----
    for (int c = tid; c < 2 * UDIM; c += 256) {
      float rb = rowbuf[c];
      for (int r = 0; r < UDIM; ++r) {
        if (r == k) continue;
        W[(size_t)r * 2 * UDIM + c] -= colbuf[r] * rb;
      }
    }
    __syncthreads();
  }

  // ---- emit Minv^T, -Minv, Minv@q (thread t owns row t of Minv) ----
  float acc = 0.0f;
  for (int c = 0; c < UDIM; ++c) {
    float m = W[(size_t)tid * 2 * UDIM + UDIM + c];
    T [(size_t)l * UDIM * UDIM + (size_t)c   * UDIM + tid] = m;  // T[k][n]=Minv[n][k]
    Nn[(size_t)l * UDIM * UDIM + (size_t)tid * UDIM + c  ] = -m;
    acc += m * q[(size_t)l * UDIM + c];
  }
  cvec[(size_t)l * UDIM + tid] = acc;
}

// ---------------------------------------------------------------------------
// fp32 WMMA GEMM:  C(M,N) = A(M,K) @ B(K,N) [+ bias[n]]; optional tanh copy.
// All operands row-major. 256 threads = 8 waves, wave grid WRxWC, each wave
// owns MTxNT tiles of 16x16 (V_WMMA_F32_16X16X4_F32).
// A staged in LDS [m][k] via async-DMA (double-buffered); B staged transposed
// [n][k] through registers so both fragments are 8B ds_load_b64 per the fp32
// 16x4 layout (lanes 0-15 -> K{0,1}, lanes 16-31 -> K{2,3}).
// Requires: M % BM == 0, K % KB == 0. Column guards used when BN == 16.
// ---------------------------------------------------------------------------
template <int BM, int BN, int WR, int WC, int MT, int NT>
__global__ __launch_bounds__(256) void wmma_gemm_f32(
    const float* __restrict__ A, const float* __restrict__ B,
    const float* __restrict__ bias, float* __restrict__ C,
    float* __restrict__ Ctanh, int M, int N, int K)
{
  static_assert(WR * WC == 8, "8 waves per workgroup");
  static_assert(WR * MT * 16 == BM && WC * NT * 16 == BN, "tile coverage");

  __shared__ float As[2][BM][KBP];
  __shared__ float Bs[2][BN][KBP];

  const int tid  = threadIdx.x;
  const int lane = tid & 31;          // wave32
  const int wave = tid >> 5;
  const int wr   = wave / WC;
  const int wc   = wave % WC;
  const int m0   = blockIdx.y * BM;
  const int n0   = blockIdx.x * BN;
  const int lr   = lane & 15;         // fragment row/col within 16
  const int h    = lane >> 4;         // K-half select (0: K{0,1}, 1: K{2,3})

  v8f acc[MT][NT] = {};

  v4f  bRegV[2];
  float bRegS = 0.0f;

  // ---- B tile: global -> regs (prefetchable), then regs -> LDS transposed
  auto load_B = [&](int kt) {
    if constexpr (BN == 128) {
      const float* Bg = B + (size_t)kt * KB * N + n0;
      #pragma unroll
      for (int i = 0; i < 2; ++i) {
        int idx  = tid + i * 256;
        int r    = idx >> 5;            // k row (0..15)
        int c    = (idx & 31) << 2;     // n col
        bRegV[i] = *(const v4f*)(Bg + (size_t)r * N + c);
      }
    } else {                            // BN == 16, guarded (N may be 10)
      int r = tid >> 4, c = tid & 15;
      int col = n0 + c;
      bRegS = (col < N) ? B[((size_t)kt * KB + r) * N + col] : 0.0f;
    }
  };
  auto store_B = [&](int buf) {
    if constexpr (BN == 128) {
      #pragma unroll
      for (int i = 0; i < 2; ++i) {
        int idx = tid + i * 256;
        int r   = idx >> 5;
        int c   = (idx & 31) << 2;
        #pragma unroll
        for (int j = 0; j < 4; ++j) Bs[buf][c + j][r] = bRegV[i][j];
      }
    } else {
      Bs[buf][tid & 15][tid >> 4] = bRegS;
    }
  };

  // ---- 4 WMMA k-steps over one staged tile
  auto compute = [&](int buf) {
    #pragma unroll
    for (int ks = 0; ks < KB / 4; ++ks) {
      v2f af[MT], bf[NT];
      #pragma unroll
      for (int mt = 0; mt < MT; ++mt)
        af[mt] = *(const v2f*)&As[buf][wr * MT * 16 + mt * 16 + lr][ks * 4 + 2 * h];
      #pragma unroll
      for (int nt = 0; nt < NT; ++nt)
        bf[nt] = *(const v2f*)&Bs[buf][wc * NT * 16 + nt * 16 + lr][ks * 4 + 2 * h];
      #pragma unroll
      for (int mt = 0; mt < MT; ++mt)
        #pragma unroll
        for (int nt = 0; nt < NT; ++nt)
          acc[mt][nt] = __builtin_amdgcn_wmma_f32_16x16x4_f32(
              false, af[mt], false, bf[nt],
              (short)0, acc[mt][nt], false, false);
    }
  };

  const int kTiles = K / KB;

#if USE_ASYNC_LDS
  // A tile (BM x KB = 2048 floats): 2 async b128 per thread, written straight
  // into LDS by the async engine (no VGPR staging), tracked by ASYNCcnt.
  auto async_load_A = [&](int kt, int buf) {
    const float* Ag = A + (size_t)m0 * K + (size_t)kt * KB;
    #pragma unroll
    for (int i = 0; i < 2; ++i) {
      int idx = tid + i * 256;
      int r   = idx >> 2;
      int c   = (idx & 3) << 2;
      __builtin_amdgcn_global_load_async_to_lds_b128(
          (AS1 b128_t*)(Ag + (size_t)r * K + c),
          (AS3 b128_t*)&As[buf][r][c],
          0, 0);
    }
  };

  async_load_A(0, 0);
  load_B(0);
  for (int kt = 0; kt < kTiles; ++kt) {
    const int cur = kt & 1;
    store_B(cur);                 // Bs[cur] free: last read in compute(kt-2),
                                  // all waves passed barrier of iter kt-1
    WAIT_ASYNC_0();               // own share of A tile kt landed in LDS
    __syncthreads();              // tile visible; all waves past compute(kt-1)
    if (kt + 1 < kTiles) {        // issue next DMA only after the barrier, so
      async_load_A(kt + 1, cur ^ 1);  // nobody still reads the target buffer
      load_B(kt + 1);
    }
    compute(cur);                 // overlaps with the in-flight async copy
  }
#else
  // Fallback: register-staged A with two barriers per tile.
  v4f aReg[2];
  auto load_A = [&](int kt) {
    const float* Ag = A + (size_t)m0 * K + (size_t)kt * KB;
    #pragma unroll
    for (int i = 0; i < 2; ++i) {
      int idx = tid + i * 256;
      int r   = idx >> 2;
      int c   = (idx & 3) << 2;
      aReg[i] = *(const v4f*)(Ag + (size_t)r * K + c);
    }
  };
  auto store_A = [&]() {
    #pragma unroll
    for (int i = 0; i < 2; ++i) {
      int idx = tid + i * 256;
      int r   = idx >> 2;
      int c   = (idx & 3) << 2;
      *(v4f*)&As[0][r][c] = aReg[i];
    }
  };

  load_A(0);
  load_B(0);
  for (int kt = 0; kt < kTiles; ++kt) {
    __syncthreads();
    store_A();
    store_B(0);
    __syncthreads();
    if (kt + 1 < kTiles) { load_A(kt + 1); load_B(kt + 1); }
    compute(0);
  }
#endif

  // ---- epilogue: C/D layout -> lanes 0-15 rows m0+r, lanes 16-31 rows m0+8+r
  #pragma unroll
  for (int mt = 0; mt < MT; ++mt) {
    #pragma unroll
    for (int nt = 0; nt < NT; ++nt) {
      int col = n0 + wc * NT * 16 + nt * 16 + lr;
      float bv = (bias != nullptr && col < N) ? bias[col] : 0.0f;
      #pragma unroll
      for (int r = 0; r < 8; ++r) {
        int row = m0 + wr * MT * 16 + mt * 16 + r + 8 * h;
        if (col < N) {
          float val = acc[mt][nt][r] + bv;
          C[(size_t)row * N + col] = val;
          if (Ctanh != nullptr) Ctanh[(size_t)row * N + col] = tanhf(val);
        }
      }
    }
  }
}

// ---------------------------------------------------------------------------
extern "C" void kernel_launch(void* const* d_in, const int* in_sizes, int n_in,
                              void* d_out, int out_size, void* d_ws, size_t ws_size,
                              hipStream_t stream)
{
  (void)in_sizes; (void)n_in; (void)out_size; (void)ws_size;

  const float* x     = (const float*)d_in[0];  // [32768,784]
  const float* W_in  = (const float*)d_in[1];  // [784,256]
  const float* b_in  = (const float*)d_in[2];  // [256]
  const float* B0    = (const float*)d_in[3];  // [6,256,256]
  const float* q     = (const float*)d_in[4];  // [6,256]
  const float* W_out = (const float*)d_in[5];  // [256,10]
  const float* b_out = (const float*)d_in[6];  // [10]
  float* out = (float*)d_out;                  // [32768,10]

  // workspace carve-up (~134 MB of fp32)
  float* p    = (float*)d_ws;
  float* T    = p; p += (size_t)LAYERS * UDIM * UDIM;      // Minv^T
  float* Nn   = p; p += (size_t)LAYERS * UDIM * UDIM;      // -Minv
  float* cvec = p; p += (size_t)LAYERS * UDIM;             // Minv @ q
  float* aug  = p; p += (size_t)LAYERS * UDIM * 2 * UDIM;  // GJ scratch
  float* V0   = p; p += (size_t)BATCH * UDIM;
  float* V1   = p; p += (size_t)BATCH * UDIM;
  float* V0b  = p; p += (size_t)BATCH * UDIM;
  float* V1b  = p; p += (size_t)BATCH * UDIM;

  // 1) invert the six 256x256 layer matrices
  gj_invert_kernel<<<LAYERS, 256, 0, stream>>>(B0, q, aug, T, Nn, cvec);

  // 2) input GEMM: V0 = x@W_in + b_in ; V1 = tanh(V0)
  dim3 gridBig(UDIM / 128, BATCH / 128);
  wmma_gemm_f32<128, 128, 2, 4, 4, 2><<<gridBig, 256, 0, stream>>>(
      x, W_in, b_in, V0, V1, BATCH, UDIM, DIN);

  // 3) six implicit layers (ping-pong):
  //    v0' = v1 @ (-Minv) ; v1' = v0 @ Minv^T + (Minv q)
  float* v0 = V0; float* v1 = V1; float* v0n = V0b; float* v1n = V1b;
  for (int l = 0; l < LAYERS; ++l) {
    wmma_gemm_f32<128, 128, 2, 4, 4, 2><<<gridBig, 256, 0, stream>>>(
        v1, Nn + (size_t)l * UDIM * UDIM, nullptr, v0n, nullptr,
        BATCH, UDIM, UDIM);
    wmma_gemm_f32<128, 128, 2, 4, 4, 2><<<gridBig, 256, 0, stream>>>(
        v0, T + (size_t)l * UDIM * UDIM, cvec + (size_t)l * UDIM, v1n, nullptr,
        BATCH, UDIM, UDIM);
    float* t;
    t = v0; v0 = v0n; v0n = t;
    t = v1; v1 = v1n; v1n = t;
  }

  // 4) output GEMM: out = v0 @ W_out + b_out  (N=10, guarded 16-wide tiles)
  dim3 gridOut(1, BATCH / 128);
  wmma_gemm_f32<128, 16, 8, 1, 1, 1><<<gridOut, 256, 0, stream>>>(
      v0, W_out, b_out, out, nullptr, BATCH, DOUT, UDIM);
}